// Aggregation_33320356282418
// MI455X (gfx1250) — compile-verified
//
#include <hip/hip_runtime.h>
#include <hip/hip_bf16.h>
#include <stdint.h>

// Aggregation (faithful raw-reshape variant):
// out[n, g*256+q, l] = sum_{r=0..8} W[n,q,r,l] * I[n, hy, wx, (s&3)*256 + l]
//   with a=q>>2, b=q&3, hw=g*64+a, hc=hw>>4, wc=hw&15,
//        s=b*9+r, p=s>>2, ki=p/3, kj=p%3, hy=hc+ki-1, wx=wc+kj-1 (zero pad OOB)
//
// HBM-bound (~142 MB total, AI ~1 flop/byte -> ~6us at 23.3 TB/s).
// CDNA5 paths: async global->LDS DMA (global_load_async_to_lds_b128 +
// s_wait_asynccnt), double-buffered over q; NT hints for the streaming
// read-once weight and the write-once output.

#define Hh 16
#define Ww 16
#define CX 1024
#define CW 256
#define K2 9
#define LL 256
#define TQ 4   // q values per block (double-buffered pipeline)

__device__ __forceinline__ void async_b128(uint32_t lds_off, uint64_t gaddr) {
    asm volatile("global_load_async_to_lds_b128 %0, %1, off"
                 :: "v"(lds_off), "v"(gaddr)
                 : "memory");
}

// Non-temporal variant for streaming (read-once) data: don't let the 75.5 MB
// weight stream evict the input's 9x-reuse working set.
__device__ __forceinline__ void async_b128_nt(uint32_t lds_off, uint64_t gaddr) {
    asm volatile("global_load_async_to_lds_b128 %0, %1, off th:TH_LOAD_NT"
                 :: "v"(lds_off), "v"(gaddr)
                 : "memory");
}

__device__ __forceinline__ uint32_t lds_addr_of(const void* p) {
    // Generic LDS pointers carry the LDS byte offset in the low 32 bits
    // (the flat->LDS aperture mapping keeps only the low 32 bits).
    return (uint32_t)(uintptr_t)p;
}

__global__ __launch_bounds__(256) void Aggregation_kernel(
    const float* __restrict__ input,   // [n,16,16,1024]
    const float* __restrict__ weight,  // [n,256,9,256]
    float* __restrict__ out)           // [n,1024,16,16]
{
    __shared__ __align__(16) float wbuf[2][K2][LL];  // 18 KB
    __shared__ __align__(16) float ibuf[2][K2][LL];  // 18 KB

    const int tid = threadIdx.x;            // = l
    const int bid = blockIdx.x;
    const int qt  = bid & 63;               // 64 q-tiles of TQ
    const int g   = (bid >> 6) & 3;
    const int n   = bid >> 8;
    const int q0  = qt * TQ;

    const float* wsrc_n = weight + (size_t)n * (CW * K2 * LL);
    const float* isrc_n = input  + (size_t)n * (Hh * Ww * CX);

    const uint32_t wlds = lds_addr_of(&wbuf[0][0][0]);
    const uint32_t ilds = lds_addr_of(&ibuf[0][0][0]);

    // ---- issue one stage of async DMA (exactly 6 instructions per wave) ----
    auto stage_load = [&](int q, int st) {
        const int a  = q >> 2, b = q & 3;
        const int hw = g * 64 + a;          // note: a == qt for all q in block
        const int hc = hw >> 4, wc = hw & 15;

        // weight tile for (n,q): 9*256 floats, contiguous 9KB -> 576 x 16B
        const float* wsrc = wsrc_n + (size_t)q * (K2 * LL);
        const uint32_t wdst = wlds + (uint32_t)st * (K2 * LL * 4);
#pragma unroll
        for (int it = 0; it < 3; ++it) {
            int idx = tid + it * 256;
            idx = (idx > 575) ? 575 : idx;   // clamp: keeps EXEC full, ASYNCcnt uniform
            async_b128_nt(wdst + (uint32_t)idx * 16, (uint64_t)(wsrc + idx * 4));
        }
        // input rows: r = idx/64, 16B chunk = idx%64 (rows disjoint across the
        // 4 stages of a block: s = b*9 + r covers [0,36) exactly once)
        const uint32_t idst = ilds + (uint32_t)st * (K2 * LL * 4);
#pragma unroll
        for (int it = 0; it < 3; ++it) {
            int idx = tid + it * 256;
            idx = (idx > 575) ? 575 : idx;
            const int r   = idx >> 6;
            const int c16 = idx & 63;
            const int s   = b * 9 + r;       // [0,36)
            const int p   = s >> 2;          // [0,9)
            const int ki  = p / 3, kj = p % 3;
            int hy = hc + ki - 1, wx = wc + kj - 1;
            // clamp OOB taps to a safe in-bounds address; compute masks them out
            hy = (hy < 0) ? 0 : ((hy > Hh - 1) ? Hh - 1 : hy);
            wx = (wx < 0) ? 0 : ((wx > Ww - 1) ? Ww - 1 : wx);
            const float* src = isrc_n + ((size_t)(hy * Ww + wx) * CX + (s & 3) * 256) + c16 * 4;
            async_b128(idst + (uint32_t)idx * 16, (uint64_t)src);
        }
    };

    // ---- consume one stage from LDS and store the output row ----
    auto compute_store = [&](int q, int st) {
        const int a  = q >> 2, b = q & 3;
        const int hw = g * 64 + a;
        const int hc = hw >> 4, wc = hw & 15;
        float acc = 0.0f;
#pragma unroll
        for (int r = 0; r < K2; ++r) {
            const int s  = b * 9 + r;
            const int p  = s >> 2;
            const int ki = p / 3, kj = p % 3;
            const int hy = hc + ki - 1, wx = wc + kj - 1;
            const bool valid = (hy >= 0) & (hy < Hh) & (wx >= 0) & (wx < Ww);
            const float wv = wbuf[st][r][tid];
            const float iv = ibuf[st][r][tid];
            if (valid) acc = fmaf(wv, iv, acc);
        }
        // write-once output, never re-read on device -> non-temporal store
        __builtin_nontemporal_store(
            acc, out + (size_t)n * (CX * LL) + (size_t)(g * CW + q) * LL + tid);
    };

    // ---- double-buffered pipeline over TQ q-values ----
    stage_load(q0, 0);
#pragma unroll
    for (int i = 0; i < TQ; ++i) {
        if (i + 1 < TQ) {
            stage_load(q0 + i + 1, (i + 1) & 1);
            // two stages in flight (12 per wave); wait until only the newest
            // stage (6) remains -> stage i fully landed in LDS (in-order done)
            asm volatile("s_wait_asynccnt 6" ::: "memory");
        } else {
            asm volatile("s_wait_asynccnt 0" ::: "memory");
        }
        __syncthreads();                 // all waves' DMA for stage i visible
        compute_store(q0 + i, i & 1);
        __syncthreads();                 // protect buf (i&1) before reuse
    }
}

extern "C" void kernel_launch(void* const* d_in, const int* in_sizes, int n_in,
                              void* d_out, int out_size, void* d_ws, size_t ws_size,
                              hipStream_t stream) {
    const float* input  = (const float*)d_in[0];   // [n,16,16,1024] f32
    const float* weight = (const float*)d_in[1];   // [n,256,9,256]  f32
    float* out = (float*)d_out;                    // [n,1024,16,16] f32

    const int n = in_sizes[0] / (Hh * Ww * CX);    // 32
    const int blocks = n * 4 * (CW / TQ);          // n * 4 * 64 = 8192
    Aggregation_kernel<<<blocks, 256, 0, stream>>>(input, weight, out);
}